// GNNModel_27427661152768
// MI455X (gfx1250) — compile-verified
//
#include <hip/hip_runtime.h>
#include <cstdint>

typedef __attribute__((ext_vector_type(2))) float v2f;
typedef __attribute__((ext_vector_type(8))) float v8f;

#define N_IN  128
#define N_HID 16
#define N_OUT 64

// ---------------------------------------------------------------------------
// 1. init: zero aggregation buffers, deg = 1.0 (self-loop contribution)
// ---------------------------------------------------------------------------
__global__ void k_init(float* __restrict__ deg, float* __restrict__ agg1,
                       float* __restrict__ agg2, int n) {
  int i = blockIdx.x * blockDim.x + threadIdx.x;
  int tot = n * N_HID;
  if (i < tot) { agg1[i] = 0.0f; agg2[i] = 0.0f; }
  if (i < n)   deg[i] = 1.0f;
}

// ---------------------------------------------------------------------------
// 2. degree: deg[col[e]] += 1  (f32 atomics, L2-resident)
// ---------------------------------------------------------------------------
__global__ void k_degree(const long long* __restrict__ col,
                         float* __restrict__ deg, long long E) {
  long long e = (long long)blockIdx.x * blockDim.x + threadIdx.x;
  if (e < E) atomicAdd(&deg[(int)col[e]], 1.0f);
}

// ---------------------------------------------------------------------------
// 3. dis = rsqrt(deg)   (deg >= 1 always because of self-loops)
// ---------------------------------------------------------------------------
__global__ void k_rsqrt(const float* __restrict__ deg, float* __restrict__ dis, int n) {
  int i = blockIdx.x * blockDim.x + threadIdx.x;
  if (i < n) dis[i] = rsqrtf(deg[i]);
}

// ---------------------------------------------------------------------------
// 4. t1 = x @ W1   [n,128]x[128,16] -> [n,16]; fp32 WMMA 16x16x4, one
//    16-row tile per wave32 (8 waves / 256-thread block -> 128 rows/block).
//    A layout (16x4 f32): lanes 0-15 M=lane, VGPR0/1 = K,K+1; lanes 16-31
//    hold K+2,K+3. B mirrored (N=lane&15, K split across lane halves).
// ---------------------------------------------------------------------------
__global__ void k_gemm1(const float* __restrict__ x, const float* __restrict__ W1,
                        float* __restrict__ t1, int n) {
  int wave = (blockIdx.x * blockDim.x + threadIdx.x) >> 5;
  int lane = threadIdx.x & 31;
  int row0 = wave * 16;
  if (row0 >= n) return;                 // wave-uniform: EXEC all-1s at WMMA
  int m    = lane & 15;
  int half = lane >> 4;                  // 0 or 1
  int row  = row0 + m; if (row > n - 1) row = n - 1;   // clamp (store-guarded)
  const float* arow = x + (size_t)row * N_IN;
  v8f acc = {};
  for (int k = 0; k < N_IN; k += 4) {
    int ka = k + half * 2;
    v2f a; a.x = arow[ka];               a.y = arow[ka + 1];
    v2f b; b.x = W1[ka * N_HID + m];     b.y = W1[(ka + 1) * N_HID + m];
    acc = __builtin_amdgcn_wmma_f32_16x16x4_f32(false, a, false, b,
                                                (short)0, acc, false, false);
  }
  // C/D layout: VGPR j -> M = j + half*8, N = lane&15
  for (int j = 0; j < 8; ++j) {
    int r = row0 + j + half * 8;
    if (r < n) t1[(size_t)r * N_HID + m] = acc[j];
  }
}

// ---------------------------------------------------------------------------
// 5. agg1[col] += t1[row] * dis[row]*dis[col]   (edges + self-loops)
//    one thread per (edge, channel): coalesced 16-float gather/scatter
// ---------------------------------------------------------------------------
__global__ void k_agg1(const long long* __restrict__ row, const long long* __restrict__ col,
                       const float* __restrict__ dis, const float* __restrict__ t1,
                       float* __restrict__ agg1, long long E, int n) {
  long long idx = (long long)blockIdx.x * blockDim.x + threadIdx.x;
  long long tot = (E + (long long)n) * N_HID;
  if (idx >= tot) return;
  long long e = idx >> 4;
  int ch = (int)(idx & 15);
  int r, c;
  if (e < E) { r = (int)row[e]; c = (int)col[e]; }
  else       { r = (int)(e - E); c = r; }          // self-loop
  float nrm = dis[r] * dis[c];
  float v = t1[(size_t)r * N_HID + ch] * nrm;
  atomicAdd(&agg1[(size_t)c * N_HID + ch], v);
}

// ---------------------------------------------------------------------------
// 6. agg2[col] += relu(agg1[row] + b1) * norm   (bias+ReLU fused into gather)
//    Layer-2 reassociation: agg(h @ W2) == agg(h) @ W2  -> 16 floats/edge.
// ---------------------------------------------------------------------------
__global__ void k_agg2(const long long* __restrict__ row, const long long* __restrict__ col,
                       const float* __restrict__ dis, const float* __restrict__ agg1,
                       const float* __restrict__ b1, float* __restrict__ agg2,
                       long long E, int n) {
  long long idx = (long long)blockIdx.x * blockDim.x + threadIdx.x;
  long long tot = (E + (long long)n) * N_HID;
  if (idx >= tot) return;
  long long e = idx >> 4;
  int ch = (int)(idx & 15);
  int r, c;
  if (e < E) { r = (int)row[e]; c = (int)col[e]; }
  else       { r = (int)(e - E); c = r; }
  float nrm = dis[r] * dis[c];
  float h = agg1[(size_t)r * N_HID + ch] + b1[ch];
  h = h > 0.0f ? h : 0.0f;
  atomicAdd(&agg2[(size_t)c * N_HID + ch], h * nrm);
}

// ---------------------------------------------------------------------------
// 7. out = sigmoid(agg2 @ W2 + b2)   [n,16]x[16,64] -> [n,64]
//    per wave: 16 rows x 64 cols = 4 accumulator tiles, 16 WMMA ops
// ---------------------------------------------------------------------------
__device__ __forceinline__ float sigmoidf_(float v) {
  return 1.0f / (1.0f + __expf(-v));
}

__global__ void k_gemm2(const float* __restrict__ agg2, const float* __restrict__ W2,
                        const float* __restrict__ bias2, float* __restrict__ out, int n) {
  int wave = (blockIdx.x * blockDim.x + threadIdx.x) >> 5;
  int lane = threadIdx.x & 31;
  int row0 = wave * 16;
  if (row0 >= n) return;                 // wave-uniform exit
  int m    = lane & 15;
  int half = lane >> 4;
  int row  = row0 + m; if (row > n - 1) row = n - 1;
  const float* arow = agg2 + (size_t)row * N_HID;
  v8f c0 = {}, c1 = {}, c2 = {}, c3 = {};
  for (int k = 0; k < N_HID; k += 4) {
    int ka = k + half * 2;
    v2f a; a.x = arow[ka]; a.y = arow[ka + 1];
    const float* w0 = W2 + (size_t)ka * N_OUT;
    const float* w1 = W2 + (size_t)(ka + 1) * N_OUT;
    v2f b;
    b.x = w0[m];          b.y = w1[m];
    c0 = __builtin_amdgcn_wmma_f32_16x16x4_f32(false, a, false, b, (short)0, c0, false, false);
    b.x = w0[16 + m];     b.y = w1[16 + m];
    c1 = __builtin_amdgcn_wmma_f32_16x16x4_f32(false, a, false, b, (short)0, c1, false, false);
    b.x = w0[32 + m];     b.y = w1[32 + m];
    c2 = __builtin_amdgcn_wmma_f32_16x16x4_f32(false, a, false, b, (short)0, c2, false, false);
    b.x = w0[48 + m];     b.y = w1[48 + m];
    c3 = __builtin_amdgcn_wmma_f32_16x16x4_f32(false, a, false, b, (short)0, c3, false, false);
  }
  for (int j = 0; j < 8; ++j) {
    int r = row0 + j + half * 8;
    if (r < n) {
      float* orow = out + (size_t)r * N_OUT;
      orow[m]      = sigmoidf_(c0[j] + bias2[m]);
      orow[16 + m] = sigmoidf_(c1[j] + bias2[16 + m]);
      orow[32 + m] = sigmoidf_(c2[j] + bias2[32 + m]);
      orow[48 + m] = sigmoidf_(c3[j] + bias2[48 + m]);
    }
  }
}

// ---------------------------------------------------------------------------
extern "C" void kernel_launch(void* const* d_in, const int* in_sizes, int n_in,
                              void* d_out, int out_size, void* d_ws, size_t ws_size,
                              hipStream_t stream) {
  const float*     x   = (const float*)d_in[0];
  const long long* ei  = (const long long*)d_in[1];
  const float*     W1  = (const float*)d_in[2];
  const float*     b1  = (const float*)d_in[3];
  const float*     W2  = (const float*)d_in[4];
  const float*     b2  = (const float*)d_in[5];
  float*           out = (float*)d_out;

  const int       n = in_sizes[0] / N_IN;          // 100000
  const long long E = (long long)in_sizes[1] / 2;  // 3200000
  const long long* row = ei;
  const long long* col = ei + E;

  // workspace layout (floats): deg[n] | dis[n] | t1[n*16] | agg1[n*16] | agg2[n*16]
  float* deg  = (float*)d_ws;
  float* dis  = deg  + n;
  float* t1   = dis  + n;
  float* agg1 = t1   + (size_t)n * N_HID;
  float* agg2 = agg1 + (size_t)n * N_HID;

  const int TB = 256;
  long long edgeWork = (E + (long long)n) * N_HID;
  int initBlocks = (n * N_HID + TB - 1) / TB;
  int degBlocks  = (int)((E + TB - 1) / TB);
  int disBlocks  = (n + TB - 1) / TB;
  int aggBlocks  = (int)((edgeWork + TB - 1) / TB);
  int gemmBlocks = (n + 127) / 128;   // 8 waves/block * 16 rows/wave

  k_init  <<<initBlocks, TB, 0, stream>>>(deg, agg1, agg2, n);
  k_degree<<<degBlocks,  TB, 0, stream>>>(col, deg, E);
  k_rsqrt <<<disBlocks,  TB, 0, stream>>>(deg, dis, n);
  k_gemm1 <<<gemmBlocks, TB, 0, stream>>>(x, W1, t1, n);
  k_agg1  <<<aggBlocks,  TB, 0, stream>>>(row, col, dis, t1, agg1, E, n);
  k_agg2  <<<aggBlocks,  TB, 0, stream>>>(row, col, dis, agg1, b1, agg2, E, n);
  k_gemm2 <<<gemmBlocks, TB, 0, stream>>>(agg2, W2, b2, out, n);
}